// CTCLossModel_50242527429436
// MI455X (gfx1250) — compile-verified
//
#include <hip/hip_runtime.h>
#include <stdint.h>

// CTC loss (Keras ctc_batch_cost semantics) for fixed shapes:
//   B=64, T=1000, C=128 (blank = C-1), L=100, S = 2L+1 = 201
//
// One wave32 per batch element; 7 extended positions per lane (224 >= 201).
// The per-timestep 128-float probability row is streamed through a 16-deep
// LDS ring buffer filled by CDNA5 async global->LDS loads (one b128 per lane
// = full 512B row per instruction). ASYNCcnt completes in order, so
// s_wait_asynccnt with a constant target proves the oldest row has landed —
// 16 rows in flight hides ~900ns of HBM latency behind ~55ns/step of VALU.
// Cross-lane alpha neighbors via shuffles; zero workgroup barriers on the
// 1000-step critical path.

namespace {
constexpr int Bn = 64;
constexpr int Tn = 1000;
constexpr int Cn = 128;
constexpr int Ln = 100;
constexpr int Sn = 2 * Ln + 1;   // 201
constexpr int PL = 7;            // extended positions per lane; 32*7 = 224
constexpr int NBUF = 16;         // async pipeline depth (power of 2)
constexpr float NEGV = -1e30f;
constexpr float EPS  = 1e-7f;
}

// CDNA5 async copy: global memory -> LDS, 16 bytes per lane.
// VDST operand = 32-bit LDS byte address, VADDR = 64-bit global address.
__device__ __forceinline__ void async_row_load_b128(uint32_t lds_addr,
                                                    const float* gaddr) {
  asm volatile("global_load_async_to_lds_b128 %0, %1, off"
               :: "v"(lds_addr), "v"(gaddr)
               : "memory");
}

// Wait until ASYNCcnt <= N (compile-time immediate). In-order completion of
// async loads means this proves the oldest (issued - N) transfers are done.
template <int N>
__device__ __forceinline__ void wait_async_le() {
  asm volatile("s_wait_asynccnt %0" :: "n"(N) : "memory");
}

__global__ __launch_bounds__(32)
void ctc_loss_wave_kernel(const int*   __restrict__ y_true,      // [B,L]
                          const float* __restrict__ y_pred,      // [B,T,C] softmax probs
                          const int*   __restrict__ in_len,      // [B]
                          const int*   __restrict__ lab_len,     // [B]
                          float*       __restrict__ out)         // [B]
{
  __shared__ float rows[NBUF][Cn];     // 16-deep ring of probability rows (8KB)
  __shared__ float afin[32 * PL];      // final alpha spill for the gather

  const int b    = blockIdx.x;
  const int lane = threadIdx.x;        // wave32
  const int len  = in_len[b];          // in [T/2, T]  (>= 500 >> NBUF)

  const float* ypb = y_pred + (size_t)b * Tn * Cn;

  // LDS byte addresses (low 32 bits of the generic pointer are the LDS offset)
  const uint32_t rows_base = (uint32_t)(uintptr_t)(&rows[0][0]);
  const uint32_t lane_off  = (uint32_t)lane * 16u;

  // ---- prologue: fill the async pipeline with rows 0 .. NBUF-1 -----------
#pragma unroll
  for (int r = 0; r < NBUF; ++r) {
    async_row_load_b128(rows_base + (uint32_t)(r * Cn * 4) + lane_off,
                        ypb + (size_t)r * Cn + lane * 4);
  }

  // ---- per-lane static tables (fully unrolled -> stay in VGPRs) ----------
  // ext[s] = blank (C-1) for even s, y_true[(s-1)/2] for odd s.
  // skip[s] = (s odd) && k>=1 && label[k] != label[k-1]   (s=1 case is covered
  //           because alpha[s-2] is NEG there anyway).
  int  eidx[PL];
  bool isblank[PL];
  bool skp[PL];
  bool valid[PL];
#pragma unroll
  for (int j = 0; j < PL; ++j) {
    const int s = lane * PL + j;
    const bool v  = (s < Sn);
    const bool bl = ((s & 1) == 0);
    int idx = Cn - 1;
    bool sk = false;
    if (v && !bl) {
      const int k  = (s - 1) >> 1;
      const int lb = y_true[b * Ln + k];
      idx = lb;
      if (k >= 1) {
        const int lbm = y_true[b * Ln + k - 1];
        sk = (lb != lbm);
      }
    }
    eidx[j]    = idx;
    isblank[j] = bl || !v;   // invalid positions read blank (in-bounds), then forced NEG
    skp[j]     = sk;
    valid[j]   = v;
  }

  // ---- t = 0 initialization (row 0 = oldest of NBUF outstanding) ---------
  wait_async_le<NBUF - 1>();
  float a[PL];
  {
    const float lpB0 = __logf(rows[0][Cn - 1] + EPS);
#pragma unroll
    for (int j = 0; j < PL; ++j) {
      const int s = lane * PL + j;
      float v = NEGV;
      if (s == 0) v = lpB0;                                  // blank
      else if (s == 1) v = __logf(rows[0][eidx[j]] + EPS);   // label[0] (lane 0 only)
      a[j] = v;
    }
  }

  // ---- one scan step (alpha update for timestep t using rowbuf) ----------
  auto step = [&](const float* __restrict__ rowbuf) {
    const float lpB = __logf(rowbuf[Cn - 1] + EPS);   // shared by all blank slots

    // neighbor alphas from lane-1 (only 2 values cross the lane boundary)
    float n1 = __shfl_up(a[PL - 1], 1, 32);   // alpha[7l - 1]
    float n2 = __shfl_up(a[PL - 2], 1, 32);   // alpha[7l - 2]
    if (lane == 0) { n1 = NEGV; n2 = NEGV; }

    float na[PL];
#pragma unroll
    for (int j = 0; j < PL; ++j) {
      const float c0 = a[j];
      const float c1 = (j == 0) ? n1 : a[j - 1];
      float       c2 = (j == 0) ? n2 : ((j == 1) ? n1 : a[j - 2]);
      c2 = skp[j] ? c2 : NEGV;

      const float lp = isblank[j] ? lpB : __logf(rowbuf[eidx[j]] + EPS);

      const float m   = fmaxf(c0, fmaxf(c1, c2));
      const float sum = __expf(c0 - m) + __expf(c1 - m) + __expf(c2 - m);
      const float v   = m + __logf(sum) + lp;
      na[j] = valid[j] ? v : NEGV;
    }
#pragma unroll
    for (int j = 0; j < PL; ++j) a[j] = na[j];
  };

  // ---- main scan: t = 1 .. len-NBUF (row t+NBUF-1 still in range) --------
  // Invariant at iteration t: rows 0..t+NBUF-2 issued (count = t+NBUF-1);
  // waiting ASYNCcnt <= NBUF-2 forces >= t+1 completions => row t resident.
  const int tmain_end = len - (NBUF - 1);
  int t = 1;
  const float* gsrc = ypb + (size_t)NBUF * Cn;            // row t+NBUF-1 source
  for (; t < tmain_end; ++t) {
    wait_async_le<NBUF - 2>();
    // refill the slot vacated by row t-1 with row t+NBUF-1
    const uint32_t dst = rows_base
                       + (uint32_t)(((t + NBUF - 1) & (NBUF - 1)) * (Cn * 4))
                       + lane_off;
    async_row_load_b128(dst, gsrc + lane * 4);
    gsrc += Cn;
    step(rows[t & (NBUF - 1)]);
  }

  // ---- tail: everything already issued; drain once, then run dry ---------
  for (; t < len; ++t) {
    wait_async_le<0>();
    step(rows[t & (NBUF - 1)]);
  }

  // ---- final gather: -logaddexp(alpha[2ll], alpha[2ll-1]) ----------------
#pragma unroll
  for (int j = 0; j < PL; ++j) afin[lane * PL + j] = a[j];
  __syncthreads();   // single-wave workgroup: degenerates, orders LDS

  if (lane == 0) {
    const int   ll = lab_len[b];
    const float x  = afin[2 * ll];
    const float y  = afin[2 * ll - 1];
    const float m  = fmaxf(x, y);
    out[b] = -(m + __logf(__expf(x - m) + __expf(y - m)));
  }
}

extern "C" void kernel_launch(void* const* d_in, const int* in_sizes, int n_in,
                              void* d_out, int out_size, void* d_ws, size_t ws_size,
                              hipStream_t stream) {
  (void)in_sizes; (void)n_in; (void)out_size; (void)d_ws; (void)ws_size;
  const int*   y_true  = (const int*)  d_in[0];   // [B,L] int32
  const float* y_pred  = (const float*)d_in[1];   // [B,T,C] float32
  const int*   in_len  = (const int*)  d_in[2];   // [B] int32
  const int*   lab_len = (const int*)  d_in[3];   // [B] int32
  float*       out     = (float*)d_out;           // [B,1] float32

  ctc_loss_wave_kernel<<<Bn, 32, 0, stream>>>(y_true, y_pred, in_len, lab_len, out);
}